// MeshGCN_42683384987841
// MI455X (gfx1250) — compile-verified
//
#include <hip/hip_runtime.h>

// ---- problem constants (match reference) -----------------------------------
#define N_NODES 100000
#define N_EDGES 3200000
#define FEAT    16
#define N_TILES (N_NODES / 16)          // 6250, exact

typedef __attribute__((ext_vector_type(2))) float v2f;
typedef __attribute__((ext_vector_type(8))) float v8f;

// ---------------------------------------------------------------------------
// fp32 atomic add that should lower to global_atomic_add_f32 on gfx1250
__device__ __forceinline__ void atomic_add_f32(float* p, float v) {
    __hip_atomic_fetch_add(p, v, __ATOMIC_RELAXED, __HIP_MEMORY_SCOPE_AGENT);
}

// ---------------------------------------------------------------------------
// deg[n] = 1.0 (self loop)
__global__ void gcn_init_deg(float* __restrict__ deg, int n) {
    int i = blockIdx.x * blockDim.x + threadIdx.x;
    if (i < n) deg[i] = 1.0f;
}

// deg[dst[e]] += 1
__global__ void gcn_count_deg(const int* __restrict__ dst,
                              float* __restrict__ deg, int nE) {
    int e = blockIdx.x * blockDim.x + threadIdx.x;
    if (e < nE) atomic_add_f32(&deg[dst[e]], 1.0f);
}

// deg -> rsqrt(deg) in place (deg >= 1 always, so no zero guard needed)
__global__ void gcn_rsqrt(float* __restrict__ deg, int n) {
    int i = blockIdx.x * blockDim.x + threadIdx.x;
    if (i < n) deg[i] = rsqrtf(deg[i]);
}

__global__ void gcn_zero(float* __restrict__ p, int n) {
    int i = blockIdx.x * blockDim.x + threadIdx.x;
    if (i < n) p[i] = 0.0f;
}

// ---------------------------------------------------------------------------
// H[tile] = X[tile] @ W  using V_WMMA_F32_16X16X4_F32, chained over K=16 (4 steps).
// One wave (32 lanes) computes one 16x16 output tile (16 nodes x 16 out-feats).
//
// A (16x4 f32):  lanes 0-15 -> M=lane,  VGPR0=K0, VGPR1=K1
//                lanes 16-31 -> M=lane-16, VGPR0=K2, VGPR1=K3
// B (4x16 f32):  lanes 0-15 hold rows K0/K1, lanes 16-31 rows K2/K3, N=lane&15
// C/D (16x16):   VGPR v: lanes 0-15 -> M=v, lanes 16-31 -> M=v+8, N=lane&15
__global__ __launch_bounds__(256) void gcn_gemm16_wmma(
        const float* __restrict__ X,   // [N, 16] row-major
        const float* __restrict__ W,   // [16, 16] row-major
        float* __restrict__ H,         // [N, 16] row-major
        int nTiles) {
    int wave = blockIdx.x * (blockDim.x >> 5) + (threadIdx.x >> 5);
    if (wave >= nTiles) return;        // whole-wave uniform exit; live waves keep EXEC all-1s

    const int lane = threadIdx.x & 31;
    const int m    = lane & 15;        // row within tile (A) / column N (B, C/D)
    const int half = lane >> 4;        // 0: K pair {0,1}; 1: K pair {2,3}
    const int base = wave * 16;

    const float* xrow = X + (size_t)(base + m) * FEAT;

    v8f acc = {};
#pragma unroll
    for (int t = 0; t < 4; ++t) {
        const int k0 = t * 4;
        const int ka = k0 + half * 2;
        v2f a, b;
        a.x = xrow[ka + 0];            // adjacent pair -> b64 load
        a.y = xrow[ka + 1];
        b.x = W[(ka + 0) * FEAT + m];
        b.y = W[(ka + 1) * FEAT + m];
        acc = __builtin_amdgcn_wmma_f32_16x16x4_f32(
                /*neg_a=*/false, a, /*neg_b=*/false, b,
                /*c_mod=*/(short)0, acc, /*reuse_a=*/false, /*reuse_b=*/false);
    }

    const int rbase = base + half * 8;
#pragma unroll
    for (int v = 0; v < 8; ++v)
        H[(size_t)(rbase + v) * FEAT + m] = acc[v];
}

// ---------------------------------------------------------------------------
// Per-edge scatter: out[dst] += H[src] * dinv[src]*dinv[dst]
// 4 threads per edge, each handles a float4 feature chunk.
__global__ void gcn_aggregate(const float* __restrict__ H,
                              const int* __restrict__ src,
                              const int* __restrict__ dst,
                              const float* __restrict__ dinv,
                              float* __restrict__ out, int nE) {
    long long tid = (long long)blockIdx.x * blockDim.x + threadIdx.x;
    int e = (int)(tid >> 2);
    if (e >= nE) return;
    int q = ((int)tid & 3) * 4;

    int s = src[e];
    int d = dst[e];
    float w = dinv[s] * dinv[d];

    const float4 hv = *(const float4*)(H + (size_t)s * FEAT + q);
    float* o = out + (size_t)d * FEAT + q;
    atomic_add_f32(o + 0, hv.x * w);
    atomic_add_f32(o + 1, hv.y * w);
    atomic_add_f32(o + 2, hv.z * w);
    atomic_add_f32(o + 3, hv.w * w);
}

// ---------------------------------------------------------------------------
// agg[n][f] = (relu?)( agg[n][f] + h[n][f]*dinv[n]^2 + b[f] )   — in place
__global__ void gcn_finalize(float* __restrict__ agg,
                             const float* __restrict__ h,
                             const float* __restrict__ dinv,
                             const float* __restrict__ b,
                             int nElem, int do_relu) {
    int i = blockIdx.x * blockDim.x + threadIdx.x;
    if (i >= nElem) return;
    int node = i >> 4;
    int f    = i & 15;
    float di = dinv[node];
    float v  = agg[i] + h[i] * (di * di) + b[f];
    if (do_relu) v = fmaxf(v, 0.0f);
    agg[i] = v;
}

// ---------------------------------------------------------------------------
extern "C" void kernel_launch(void* const* d_in, const int* in_sizes, int n_in,
                              void* d_out, int out_size, void* d_ws, size_t ws_size,
                              hipStream_t stream) {
    const float* x  = (const float*)d_in[0];
    const int*   ei = (const int*)  d_in[1];   // [2, E]
    const float* W1 = (const float*)d_in[2];
    const float* b1 = (const float*)d_in[3];
    const float* W2 = (const float*)d_in[4];
    const float* b2 = (const float*)d_in[5];
    float* out = (float*)d_out;

    const int* src = ei;
    const int* dst = ei + N_EDGES;

    // workspace layout (floats): dinv | h | agg   (~13.3 MB total)
    float* dinv = (float*)d_ws;
    float* h    = dinv + ((N_NODES + 255) & ~255);
    float* agg  = h + (size_t)N_NODES * FEAT;

    const int T = 256;
    const int nElem = N_NODES * FEAT;
    const int gN    = (N_NODES + T - 1) / T;
    const int gE    = (N_EDGES + T - 1) / T;
    const int gE4   = (int)(((long long)N_EDGES * 4 + T - 1) / T);
    const int gElem = (nElem + T - 1) / T;
    const int gGemm = (N_TILES + (T / 32) - 1) / (T / 32);

    // degrees -> dinv
    gcn_init_deg <<<gN,  T, 0, stream>>>(dinv, N_NODES);
    gcn_count_deg<<<gE,  T, 0, stream>>>(dst, dinv, N_EDGES);
    gcn_rsqrt    <<<gN,  T, 0, stream>>>(dinv, N_NODES);

    // ---- layer 1: h = x@W1 ; agg = scatter ; agg = relu(agg + self + b1)
    gcn_gemm16_wmma<<<gGemm, T, 0, stream>>>(x, W1, h, N_TILES);
    gcn_zero       <<<gElem, T, 0, stream>>>(agg, nElem);
    gcn_aggregate  <<<gE4,   T, 0, stream>>>(h, src, dst, dinv, agg, N_EDGES);
    gcn_finalize   <<<gElem, T, 0, stream>>>(agg, h, dinv, b1, nElem, 1);

    // ---- layer 2: h = agg@W2 ; out = scatter ; out = out + self + b2
    gcn_gemm16_wmma<<<gGemm, T, 0, stream>>>(agg, W2, h, N_TILES);
    gcn_zero       <<<gElem, T, 0, stream>>>(out, nElem);
    gcn_aggregate  <<<gE4,   T, 0, stream>>>(h, src, dst, dinv, out, N_EDGES);
    gcn_finalize   <<<gElem, T, 0, stream>>>(out, h, dinv, b2, nElem, 0);
}